// StructuralAnalyzer_40226663694512
// MI455X (gfx1250) — compile-verified
//
#include <hip/hip_runtime.h>
#include <math.h>

// MI455X / gfx1250, wave32. Fused structural analyzer:
//  - one global read of mag/ang, one global write of (entropy,alignment,curvature)
//  - all intermediates in LDS (~48KB/WG of the 320KB/WGP budget)
//  - vertical Gaussian passes as banded matmul on V_WMMA_F32_16X16X4_F32
//    (fp32 matrix pipe co-executes with v_sin/v_cos/v_log/v_rcp on the VALU)
//  - raw HW rcp/sqrt/log2 (1-ulp) instead of guarded device-lib sequences;
//    entropy uses log2 natively: -(p1*ln p1 + p2*ln p2)/ln2 == -(p1*log2 p1 + p2*log2 p2)
//  - double-buffered horizontal-pass scratch: one barrier per conv field

typedef float v2f __attribute__((ext_vector_type(2)));
typedef float v8f __attribute__((ext_vector_type(8)));

#define IMG_H 512
#define IMG_W 512
#define TS    32      // output tile per workgroup (32x32), 2x2 wave tiles of 16x16
#define HALO  3       // sobel(1) + gauss-of-curvature(2); covers gauss(2) too
#define LDP   40      // LDS row pitch (floats)
#define NT    128     // 4 wave32 waves per workgroup

__device__ __forceinline__ float fast_rcp(float x)  { return __builtin_amdgcn_rcpf(x); }
__device__ __forceinline__ float fast_sqrt(float x) { return __builtin_amdgcn_sqrtf(x); }
__device__ __forceinline__ float fast_log2(float x) { return __builtin_amdgcn_logf(x); } // v_log_f32 = log2

__global__ __launch_bounds__(NT)
void structural_analyzer_kernel(const float* __restrict__ vf, float* __restrict__ out)
{
    // separable Gaussian taps, sigma=1, window 5 (normalized; product == 2D kernel)
    const float g0 = 0.0544886845f, g1 = 0.2442013420f, g2 = 0.4026199469f;
    const float gk[5] = {g0, g1, g2, g1, g0};

    __shared__ float P0[38*LDP]; // cos field
    __shared__ float P1[38*LDP]; // sin field
    __shared__ float P2[38*LDP]; // vx*vx
    __shared__ float P3[38*LDP]; // vy*vy
    __shared__ float P4[38*LDP]; // vx*vy
    __shared__ float HTa[36*LDP]; // horizontal-pass scratch, double buffered
    __shared__ float HTb[36*LDP];
    __shared__ float CV[36*LDP]; // curvature field (36x36 used)

    const int tid  = threadIdx.x;
    const int lane = tid & 31;
    const int wid  = tid >> 5;
    const int hig  = lane >> 4;     // lane half: 0 or 1
    const int lo16 = lane & 15;
    const int wy   = (wid >> 1) * 16;   // wave tile origin inside block tile
    const int wx   = (wid & 1) * 16;

    const int tx0 = blockIdx.x * TS;
    const int ty0 = blockIdx.y * TS;
    const int bb  = blockIdx.z;

    // ---- constant A fragments: banded Gaussian matrix G(16x20), K in 5 chunks of 4.
    // f32 16x4 A layout (ISA 7.12.2): M = lane&15 ; K = vgpr + 2*(lane>=16).
    // A[m][K] = g[K-m] for 0 <= K-m <= 4, else 0.  Same matrix for every conv.
    v2f afrag[5];
#pragma unroll
    for (int c = 0; c < 5; ++c) {
        int Kx = 4*c + 2*hig;
        int dx = Kx - lo16, dy = dx + 1;
        afrag[c].x = (dx >= 0 && dx <= 4) ? gk[dx] : 0.0f;
        afrag[c].y = (dy >= 0 && dy <= 4) ? gk[dy] : 0.0f;
    }

    v8f entAcc = {}, aliAcc = {}, curAcc = {};

    for (int ci = 0; ci < 4; ++ci) {
        const float* mag = vf + (size_t)(bb*8 + ci)     * IMG_H * IMG_W;
        const float* ang = vf + (size_t)(bb*8 + 4 + ci) * IMG_H * IMG_W;

        __syncthreads(); // previous channel's LDS readers done before we rewrite planes

        // ---- Phase A: halo load (38x38) + pointwise fields.
        // Zero-padding semantics: fields (cos/sin/products) are ZERO outside the
        // image (JAX pads the convolved field with zeros, not the angle).
        for (int idx = tid; idx < 38*38; idx += NT) {
            int ly = idx / 38, lx = idx - 38*ly;
            int ys = ty0 - HALO + ly, xs = tx0 - HALO + lx;
            bool inb = (ys >= 0) & (ys < IMG_H) & (xs >= 0) & (xs < IMG_W);
            float m = 0.f, a = 0.f;
            if (inb) { size_t o = (size_t)ys * IMG_W + xs; m = mag[o]; a = ang[o]; }
            float sn, cs;
            __sincosf(a, &sn, &cs);
            if (!inb) { sn = 0.f; cs = 0.f; }
            float vx = m * cs, vy = m * sn;
            int p = ly*LDP + lx;
            P0[p] = cs; P1[p] = sn; P2[p] = vx*vx; P3[p] = vy*vy; P4[p] = vx*vy;
        }
        if (ci + 1 < 4) { // pull next channel's tile toward the caches (global_prefetch_b8)
            const float* nm = vf + (size_t)(bb*8 + ci + 1) * IMG_H * IMG_W;
            int py = ty0 + (tid >> 2), px = tx0 + ((tid & 3) << 3);
            if (py < IMG_H && px < IMG_W) __builtin_prefetch(&nm[(size_t)py*IMG_W + px], 0, 0);
        }
        __syncthreads();

        // ---- Phase B: 5 separable Gaussian convs (cos,sin,vxx,vyy,vxy).
        // Horizontal taps on VALU into HT (36 rows x 32 cols, ping-pong buffers),
        // vertical pass as banded fp32 WMMA: Out(16x16) = Sum_c A(16x4) x B(4x16).
        v8f conv5[5];
        const float* planes[5] = {P0, P1, P2, P3, P4};
        float* hbuf[2] = {HTa, HTb};
#pragma unroll
        for (int f = 0; f < 5; ++f) {
            const float* Pf = planes[f];
            float* HT = hbuf[f & 1];
            for (int idx = tid; idx < 36*32; idx += NT) { // exactly 9 iters/thread
                int hy = idx >> 5, hx = idx & 31;
                int b = (hy + 1)*LDP + (hx + 1);
                HT[hy*LDP + hx] = g0*(Pf[b] + Pf[b+4]) + g1*(Pf[b+1] + Pf[b+3]) + g2*Pf[b+2];
            }
            __syncthreads(); // write -> read; ping-pong protects read vs next-next write
            v8f acc = {};
#pragma unroll
            for (int c = 0; c < 5; ++c) {
                int Kx = 4*c + 2*hig;
                // f32 4x16 B layout (symmetric w/ A): N = lane&15, K = vgpr + 2*(lane>=16)
                v2f bfrag;
                bfrag.x = HT[(wy + Kx    )*LDP + wx + lo16];
                bfrag.y = HT[(wy + Kx + 1)*LDP + wx + lo16];
                acc = __builtin_amdgcn_wmma_f32_16x16x4_f32(
                        false, afrag[c], false, bfrag, (short)0, acc, false, false);
            }
            conv5[f] = acc;   // f: 0=ux_avg 1=uy_avg 2=vxx_s 3=vyy_s 4=vxy_s
        }
        __syncthreads(); // last HT reads done before curvature phase reuses buffers

        // ---- Phase C: Sobel -> curvature field on 36x36 (zero outside image).
        for (int idx = tid; idx < 36*36; idx += NT) {
            int hy = idx / 36, hx = idx - 36*hy;
            int ys = ty0 - 2 + hy, xs = tx0 - 2 + hx;
            bool inb = (ys >= 0) & (ys < IMG_H) & (xs >= 0) & (xs < IMG_W);
            int r0 = hy*LDP + hx, r1 = r0 + LDP, r2 = r1 + LDP; // center = P[hy+1][hx+1]
            float c00=P0[r0], c01=P0[r0+1], c02=P0[r0+2];
            float c10=P0[r1],               c12=P0[r1+2];
            float c20=P0[r2], c21=P0[r2+1], c22=P0[r2+2];
            float s00=P1[r0], s01=P1[r0+1], s02=P1[r0+2];
            float s10=P1[r1],               s12=P1[r1+2];
            float s20=P1[r2], s21=P1[r2+1], s22=P1[r2+2];
            float dudx = ((c02-c00) + 2.f*(c12-c10) + (c22-c20)) * 0.125f;
            float dudy = ((c20+2.f*c21+c22) - (c00+2.f*c01+c02)) * 0.125f;
            float dvdx = ((s02-s00) + 2.f*(s12-s10) + (s22-s20)) * 0.125f;
            float dvdy = ((s20+2.f*s21+s22) - (s00+2.f*s01+s02)) * 0.125f;
            float cv = fast_sqrt(dudx*dudx + dudy*dudy + dvdx*dvdx + dvdy*dvdy + 1e-10f);
            CV[hy*LDP + hx] = inb ? cv : 0.f;
        }
        __syncthreads();

        // Gaussian of curvature: horizontal taps then banded WMMA vertical.
        for (int idx = tid; idx < 36*32; idx += NT) {
            int hy = idx >> 5, hx = idx & 31;
            int b = hy*LDP + hx;
            HTa[hy*LDP + hx] = g0*(CV[b] + CV[b+4]) + g1*(CV[b+1] + CV[b+3]) + g2*CV[b+2];
        }
        __syncthreads();
        v8f curvS = {};
#pragma unroll
        for (int c = 0; c < 5; ++c) {
            int Kx = 4*c + 2*hig;
            v2f bfrag;
            bfrag.x = HTa[(wy + Kx    )*LDP + wx + lo16];
            bfrag.y = HTa[(wy + Kx + 1)*LDP + wx + lo16];
            curvS = __builtin_amdgcn_wmma_f32_16x16x4_f32(
                      false, afrag[c], false, bfrag, (short)0, curvS, false, false);
        }

        // ---- Phase D: per-pixel nonlinear stats in the C-fragment registers.
        // Raw v_rcp/v_sqrt/v_log (1-ulp). Entropy in base-2 directly:
        //   -(p1*ln(p1)+p2*ln(p2))/ln2 == -(p1*log2(p1)+p2*log2(p2)).
        // Arguments are >= 1e-10, far above denormals, so no guards needed.
        const float EPSC = 1e-10f;
#pragma unroll
        for (int r = 0; r < 8; ++r) {
            float xx = conv5[2][r], yy = conv5[3][r], xy = conv5[4][r];
            float tr = xx + yy;
            float det = xx*yy - xy*xy;
            float disc = fast_sqrt(fmaxf(tr*tr - 4.f*det, 0.f) + EPSC);
            float l1 = fmaxf((tr + disc)*0.5f, EPSC);
            float l2 = fmaxf((tr - disc)*0.5f, EPSC);
            float rs = fast_rcp(l1 + l2 + EPSC);
            float p1 = l1 * rs, p2 = l2 * rs;
            float ent = -(p1*fast_log2(p1 + EPSC) + p2*fast_log2(p2 + EPSC));
            entAcc[r] += fminf(fmaxf(ent, 0.f), 1.f);
            float ux = conv5[0][r], uy = conv5[1][r];
            aliAcc[r] += fminf(fast_sqrt(ux*ux + uy*uy + EPSC), 1.f);
            curAcc[r] += curvS[r];
        }
    }

    // ---- write means: out[b][plane][0][y][x], plane = 0 ent / 1 ali / 2 curv.
    // C/D layout: VGPR r -> M = r + 8*(lane>=16), N = lane&15.
#pragma unroll
    for (int r = 0; r < 8; ++r) {
        int y = ty0 + wy + r + 8*hig;
        int x = tx0 + wx + lo16;
        size_t pix = (size_t)y * IMG_W + x;
        size_t plane = (size_t)IMG_H * IMG_W;
        out[((size_t)bb*3 + 0)*plane + pix] = entAcc[r] * 0.25f;
        out[((size_t)bb*3 + 1)*plane + pix] = aliAcc[r] * 0.25f;
        out[((size_t)bb*3 + 2)*plane + pix] = curAcc[r] * 0.25f;
    }
}

extern "C" void kernel_launch(void* const* d_in, const int* in_sizes, int n_in,
                              void* d_out, int out_size, void* d_ws, size_t ws_size,
                              hipStream_t stream) {
    const float* vf = (const float*)d_in[0];
    float* out = (float*)d_out;
    dim3 grid(IMG_W / TS, IMG_H / TS, 8);   // 16 x 16 x 8 = 2048 workgroups
    structural_analyzer_kernel<<<grid, NT, 0, stream>>>(vf, out);
}